// GCN_12249246728930
// MI455X (gfx1250) — compile-verified
//
#include <hip/hip_runtime.h>

#define NN   50000
#define EE   800000
#define HHC  64
#define GGC  500
#define CCC  6
#define EPSV 1e-5f

typedef float v2f __attribute__((ext_vector_type(2)));
typedef float v8f __attribute__((ext_vector_type(8)));

__device__ __forceinline__ void atomAddF(float* p, float v) {
  unsafeAtomicAdd(p, v);   // lowers to global_atomic_add_f32 on gfx1250
}

// ---------------------------------------------------------------- utilities
__global__ __launch_bounds__(256) void k_fill(float* __restrict__ p, int n, float v) {
  int t = blockIdx.x * 256 + threadIdx.x;
  if (t < n) p[t] = v;
}

__global__ __launch_bounds__(256) void k_degree(const int* __restrict__ dstIdx,
                                                float* __restrict__ deg) {
  int e = blockIdx.x * 256 + threadIdx.x;
  if (e < EE) atomAddF(&deg[dstIdx[e]], 1.0f);
}

__global__ __launch_bounds__(256) void k_rsqrt_inplace(float* __restrict__ d, int n) {
  int t = blockIdx.x * 256 + threadIdx.x;
  if (t < n) d[t] = rsqrtf(d[t]);   // deg >= 1 (self-loops)
}

// ------------------------------------------------- fp32 WMMA GEMM  Y = X * W
// X: [nrows x 64], W: [64 x 64], Y: [nrows x 64].  nrows multiple of 16.
// One wave -> one 16-row M-tile, full 64-wide N (4 accumulators), K in 16
// steps of 4 via V_WMMA_F32_16X16X4_F32 (exact fp32, matches reference).
#define WT_STRIDE 68   // 64 + 4 pad: lane bank = 4*lane -> conflict-free b64

__global__ __launch_bounds__(256) void k_gemm64_wmma(const float* __restrict__ X,
                                                     const float* __restrict__ W,
                                                     float* __restrict__ Y,
                                                     int nrows) {
  __shared__ float Wt[64 * WT_STRIDE];   // transposed: Wt[n*68 + k] = W[k][n]
  const int tid = threadIdx.x;
  for (int i = tid; i < 64 * 64; i += 256) {
    int k = i >> 6, n = i & 63;
    Wt[n * WT_STRIDE + k] = W[i];
  }
  __syncthreads();

  const int wave  = tid >> 5;                 // 8 waves / block
  const int lane  = tid & 31;
  const int mtile = blockIdx.x * 8 + wave;
  if (mtile * 16 >= nrows) return;            // wave-uniform: EXEC stays all-1s

  const int half = lane >> 4;                 // 0: lanes 0-15, 1: lanes 16-31
  const int l15  = lane & 15;
  const int koff = half * 2;                  // A/B K sub-offset per half-wave
  const float* xrow = X + (size_t)(mtile * 16 + l15) * HHC + koff;
  const float* wt0  = &Wt[(0 * 16 + l15) * WT_STRIDE + koff];
  const float* wt1  = &Wt[(1 * 16 + l15) * WT_STRIDE + koff];
  const float* wt2  = &Wt[(2 * 16 + l15) * WT_STRIDE + koff];
  const float* wt3  = &Wt[(3 * 16 + l15) * WT_STRIDE + koff];

  v8f c0 = {}, c1 = {}, c2 = {}, c3 = {};
#pragma unroll
  for (int kk = 0; kk < 16; ++kk) {
    const int kb = kk * 4;
    v2f a  = *(const v2f*)(xrow + kb);         // A 16x4 fragment (2 VGPRs)
    v2f b0 = *(const v2f*)(wt0 + kb);          // B 4x16 fragments from LDS
    v2f b1 = *(const v2f*)(wt1 + kb);
    v2f b2 = *(const v2f*)(wt2 + kb);
    v2f b3 = *(const v2f*)(wt3 + kb);
    c0 = __builtin_amdgcn_wmma_f32_16x16x4_f32(false, a, false, b0, (short)0, c0, false, false);
    c1 = __builtin_amdgcn_wmma_f32_16x16x4_f32(false, a, false, b1, (short)0, c1, false, false);
    c2 = __builtin_amdgcn_wmma_f32_16x16x4_f32(false, a, false, b2, (short)0, c2, false, false);
    c3 = __builtin_amdgcn_wmma_f32_16x16x4_f32(false, a, false, b3, (short)0, c3, false, false);
  }

  float* yb = Y + (size_t)mtile * 16 * HHC;
#pragma unroll
  for (int r = 0; r < 8; ++r) {
    int orow = half ? (8 + r) : r;             // C/D layout: VGPR r -> M=r / M=8+r
    float* yr = yb + orow * HHC + l15;
    yr[0]  = c0[r];
    yr[16] = c1[r];
    yr[32] = c2[r];
    yr[48] = c3[r];
  }
}

// ------------------------------------- aggregation: agg = D^-1/2 A D^-1/2 hlin
// Self-loop term doubles as the zero-init of agg (one fewer pass).
__global__ __launch_bounds__(256) void k_selfloop_init(const float* __restrict__ hlin,
                                                       const float* __restrict__ dinv,
                                                       float* __restrict__ agg) {
  int t = blockIdx.x * 256 + threadIdx.x;      // NN*16 threads, 4 feats each
  int n = t >> 4;
  if (n >= NN) return;
  int f = (t & 15) * 4;
  float w = dinv[n]; w *= w;
  const float4 v = *(const float4*)(hlin + (size_t)n * HHC + f);
  *(float4*)(agg + (size_t)n * HHC + f) = make_float4(w * v.x, w * v.y, w * v.z, w * v.w);
}

__global__ __launch_bounds__(256) void k_scatter(const int* __restrict__ src,
                                                 const int* __restrict__ dst,
                                                 const float* __restrict__ dinv,
                                                 const float* __restrict__ hlin,
                                                 float* __restrict__ agg) {
  long long t = (long long)blockIdx.x * 256 + threadIdx.x;  // EE*16 threads
  int e = (int)(t >> 4);
  if (e >= EE) return;
  int f = ((int)t & 15) * 4;
  int s = src[e], d = dst[e];
  float w = dinv[s] * dinv[d];
  const float4 v = *(const float4*)(hlin + (size_t)s * HHC + f);
  float* o = agg + (size_t)d * HHC + f;
  atomAddF(o + 0, w * v.x);
  atomAddF(o + 1, w * v.y);
  atomAddF(o + 2, w * v.z);
  atomAddF(o + 3, w * v.w);
}

// ------------------------------------------------- BatchNorm stats + apply
__global__ __launch_bounds__(256) void k_stats(const float* __restrict__ H,
                                               float* __restrict__ stats, int nrows) {
  __shared__ float ssum[256], ssq[256];
  const int c = threadIdx.x & 63;
  float s = 0.f, q = 0.f;
  for (int r = blockIdx.x * 4 + (threadIdx.x >> 6); r < nrows; r += gridDim.x * 4) {
    float v = H[(size_t)r * HHC + c];
    s += v; q += v * v;
  }
  ssum[threadIdx.x] = s; ssq[threadIdx.x] = q;
  __syncthreads();
  if (threadIdx.x < 64) {
    s = ssum[threadIdx.x] + ssum[threadIdx.x + 64] + ssum[threadIdx.x + 128] + ssum[threadIdx.x + 192];
    q = ssq [threadIdx.x] + ssq [threadIdx.x + 64] + ssq [threadIdx.x + 128] + ssq [threadIdx.x + 192];
    atomAddF(&stats[c], s);
    atomAddF(&stats[64 + c], q);
  }
}

// note: conv bias b cancels exactly inside batch-stat BN -> intentionally skipped
__global__ __launch_bounds__(256) void k_bnrelu(float* __restrict__ H,
                                                const float* __restrict__ stats,
                                                const float* __restrict__ gamma,
                                                const float* __restrict__ beta,
                                                float inv_n) {
  int t = blockIdx.x * 256 + threadIdx.x;      // NN*64 threads
  if (t >= NN * HHC) return;
  int c = t & 63;
  float mu  = stats[c] * inv_n;
  float var = stats[64 + c] * inv_n - mu * mu;
  float sc  = gamma[c] * rsqrtf(var + EPSV);
  float v   = (H[t] - mu) * sc + beta[c];
  H[t] = fmaxf(v, 0.0f);
}

// ------------------------------------------------- mean pool + classifier
__global__ __launch_bounds__(256) void k_pool(const float* __restrict__ H,
                                              const int* __restrict__ batch,
                                              float* __restrict__ pooled,
                                              float* __restrict__ counts) {
  int t = blockIdx.x * 256 + threadIdx.x;      // NN*16 threads
  int n = t >> 4;
  if (n >= NN) return;
  int f = (t & 15) * 4;
  int b = batch[n];
  const float4 v = *(const float4*)(H + (size_t)n * HHC + f);
  float* o = pooled + (size_t)b * HHC + f;
  atomAddF(o + 0, v.x);
  atomAddF(o + 1, v.y);
  atomAddF(o + 2, v.z);
  atomAddF(o + 3, v.w);
  if ((t & 15) == 0) atomAddF(&counts[b], 1.0f);
}

__global__ __launch_bounds__(256) void k_fc(const float* __restrict__ pooled,
                                            const float* __restrict__ counts,
                                            const float* __restrict__ fcW,
                                            const float* __restrict__ fcb,
                                            float* __restrict__ out) {
  int t = blockIdx.x * 256 + threadIdx.x;
  if (t >= GGC * CCC) return;
  int g = t / CCC, c = t % CCC;
  float inv = 1.0f / fmaxf(counts[g], 1.0f);   // mean commutes with linear map
  const float* p = pooled + (size_t)g * HHC;
  float s = 0.f;
#pragma unroll
  for (int k = 0; k < HHC; ++k) s += p[k] * fcW[k * CCC + c];
  out[t] = s * inv + fcb[c];
}

// ---------------------------------------------------------------- launcher
extern "C" void kernel_launch(void* const* d_in, const int* in_sizes, int n_in,
                              void* d_out, int out_size, void* d_ws, size_t ws_size,
                              hipStream_t stream) {
  (void)in_sizes; (void)n_in; (void)out_size; (void)ws_size;
  const float* x     = (const float*)d_in[0];
  const int*   ei    = (const int*)  d_in[1];
  const int*   batch = (const int*)  d_in[2];
  const float* W[3]  = {(const float*)d_in[3],  (const float*)d_in[7],  (const float*)d_in[11]};
  const float* ga[3] = {(const float*)d_in[5],  (const float*)d_in[9],  (const float*)d_in[13]};
  const float* be[3] = {(const float*)d_in[6],  (const float*)d_in[10], (const float*)d_in[14]};
  const float* fcW   = (const float*)d_in[15];
  const float* fcb   = (const float*)d_in[16];
  float*       out   = (float*)d_out;

  float* ws     = (float*)d_ws;
  float* buf0   = ws;                              // GEMM output (hlin)
  float* buf1   = buf0 + (size_t)NN * HHC;         // aggregation / normalized h
  float* dinv   = buf1 + (size_t)NN * HHC;         // degree -> rsqrt(degree)
  float* stats  = dinv + NN;                       // 64 sums + 64 sumsq
  float* pooled = stats + 128;                     // [G x 64]
  float* counts = pooled + (size_t)GGC * HHC;      // [G]

  const int* src = ei;        // edge_index[0]
  const int* dst = ei + EE;   // edge_index[1]

  // degrees (self-loop => init 1), then dinv = rsqrt(deg)
  k_fill         <<<(NN + 255) / 256, 256, 0, stream>>>(dinv, NN, 1.0f);
  k_degree       <<<(EE + 255) / 256, 256, 0, stream>>>(dst, dinv);
  k_rsqrt_inplace<<<(NN + 255) / 256, 256, 0, stream>>>(dinv, NN);

  const int gemmBlocks = (NN / 16 + 7) / 8;        // 3125 M-tiles, 8 waves/block
  const float* cur = x;
  for (int l = 0; l < 3; ++l) {
    k_gemm64_wmma  <<<gemmBlocks, 256, 0, stream>>>(cur, W[l], buf0, NN);
    k_selfloop_init<<<(NN * 16 + 255) / 256, 256, 0, stream>>>(buf0, dinv, buf1);
    k_scatter      <<<(EE * 16 + 255) / 256, 256, 0, stream>>>(src, dst, dinv, buf0, buf1);
    k_fill         <<<1, 128, 0, stream>>>(stats, 128, 0.0f);
    k_stats        <<<512, 256, 0, stream>>>(buf1, stats, NN);
    k_bnrelu       <<<(NN * HHC + 255) / 256, 256, 0, stream>>>(buf1, stats, ga[l], be[l],
                                                                1.0f / (float)NN);
    cur = buf1;
  }

  k_fill<<<(GGC * HHC + GGC + 255) / 256, 256, 0, stream>>>(pooled, GGC * HHC + GGC, 0.0f);
  k_pool<<<(NN * 16 + 255) / 256, 256, 0, stream>>>(buf1, batch, pooled, counts);
  k_fc  <<<(GGC * CCC + 255) / 256, 256, 0, stream>>>(pooled, counts, fcW, fcb, out);
}